// SpectralConv3d_15075335209172
// MI455X (gfx1250) — compile-verified
//
#include <hip/hip_runtime.h>

// ---------------------------------------------------------------------------
// SpectralConv3d (FNO layer) for gfx1250 via V_WMMA_F32_16X16X4_F32.
// B=2, CI=CO=32, N=64^3, modes 8^3 in 4 x/y corners (kz low 8 only).
// All operand matrices are prebuilt in LDS:
//   A (MxK): row-major, k contiguous  -> fragment = one ds_load_b64
//   B (KxN): even/odd-k interleaved [((k>>1)*N+n)*2+(k&1)] -> one ds_load_b64
// ---------------------------------------------------------------------------

typedef float v2f __attribute__((ext_vector_type(2)));
typedef float v8f __attribute__((ext_vector_type(8)));

__device__ __forceinline__ v8f wmma4(v2f a, v2f b, v8f c) {
  return __builtin_amdgcn_wmma_f32_16x16x4_f32(false, a, false, b, (short)0, c,
                                               false, false);
}

__device__ __forceinline__ v2f ld2(const float* p) { return *(const v2f*)p; }

#define STEP_2PI_64 0.09817477042468103869f  // 2*pi/64

__device__ __forceinline__ void init_tw(float* cs, float* sn, int tid) {
  if (tid < 64) {
    float s, c;
    __sincosf(STEP_2PI_64 * (float)tid, &s, &c);
    sn[tid] = s;
    cs[tid] = c;
  }
}

// ===========================================================================
// K1: per (b,ci,X): z-DFT (kz 0..7) then y-DFT (16 corner ky modes).
//   t2[bci,X, ky(16), kz(8), {r,i}]  (256 floats per (bci,X))
// ===========================================================================
__global__ __launch_bounds__(128) void k1_fwd_yz(const float* __restrict__ x,
                                                 float* __restrict__ t2) {
  __shared__ __attribute__((aligned(16))) float cs[64], sn[64];
  __shared__ __attribute__((aligned(16))) float xt[64 * 68];  // x[Y][Z]
  __shared__ __attribute__((aligned(16))) float ez[1024];  // Ez 64x16 intlv
  __shared__ __attribute__((aligned(16))) float ay[2048];  // A' 16x128 rowmaj
  __shared__ __attribute__((aligned(16))) float bt[2048];  // B' 128x16 intlv
  __shared__ __attribute__((aligned(16))) float pt[4 * 256];

  const int tid = threadIdx.x;
  const int lane = tid & 31;
  const int wave = tid >> 5;
  const int ml = lane & 15;
  const int kh = lane >> 4;

  init_tw(cs, sn, tid);
  const float* xp = x + (size_t)blockIdx.x * 4096;  // blockIdx = bci*64 + X
  for (int i = 0; i < 32; ++i) {
    int e = tid + i * 128;
    xt[(e >> 6) * 68 + (e & 63)] = xp[e];
  }
  __syncthreads();

  // Build Ez (B of stage 1): Ez[z, n] = n<8 ? cos(z*n) : -sin(z*(n-8))
  for (int i = 0; i < 8; ++i) {
    int e = tid + i * 128;  // 1024
    int n = (e >> 1) & 15, k = (e >> 5) * 2 + (e & 1);
    ez[e] = (n < 8) ? cs[(k * n) & 63] : -sn[(k * (n - 8)) & 63];
  }
  // Build A' of stage 2: ay[ky, k'] ; k'<64 -> cos(y*ky), else sin(y*ky)
  for (int i = 0; i < 16; ++i) {
    int e = tid + i * 128;  // 2048
    int m = e >> 7, k = e & 127;
    int kyact = (m < 8) ? m : (m + 48);
    ay[e] = (k < 64) ? cs[(kyact * k) & 63] : sn[(kyact * (k - 64)) & 63];
  }
  __syncthreads();

  // ---- stage 1: T[Y,n] = sum_z x[Y,z]*Ez[z,n]; write straight into the
  //      interleaved stage-2 B': rows y -> [Tr|Ti], rows 64+y -> [Ti|-Tr]
  {
    const int y0 = wave * 16;
    v8f acc = {0.f, 0.f, 0.f, 0.f, 0.f, 0.f, 0.f, 0.f};
#pragma unroll
    for (int k0 = 0; k0 < 64; k0 += 4) {
      const int kk = k0 + 2 * kh;
      v2f a = ld2(&xt[(y0 + ml) * 68 + kk]);
      v2f b = ld2(&ez[((kk >> 1) * 16 + ml) * 2]);
      acc = wmma4(a, b, acc);
    }
    const int col2 = (ml < 8) ? (ml + 8) : (ml - 8);
    const float sgn2 = (ml < 8) ? -1.f : 1.f;
#pragma unroll
    for (int v = 0; v < 8; ++v) {
      const int y = y0 + v + 8 * kh;
      const float val = acc[v];
      bt[((y >> 1) * 16 + ml) * 2 + (y & 1)] = val;                   // k=y
      bt[((32 + (y >> 1)) * 16 + col2) * 2 + (y & 1)] = sgn2 * val;   // k=64+y
    }
  }
  __syncthreads();

  // ---- stage 2: t2[ky,n] = sum_{k'} ay[ky,k'] * bt[k',n], split-K x4 waves
  {
    v8f acc = {0.f, 0.f, 0.f, 0.f, 0.f, 0.f, 0.f, 0.f};
    const int kbeg = wave * 32;
#pragma unroll
    for (int k0 = 0; k0 < 32; k0 += 4) {
      const int kk = kbeg + k0 + 2 * kh;
      v2f a = ld2(&ay[ml * 128 + kk]);
      v2f b = ld2(&bt[((kk >> 1) * 16 + ml) * 2]);
      acc = wmma4(a, b, acc);
    }
#pragma unroll
    for (int v = 0; v < 8; ++v)
      pt[wave * 256 + (v + 8 * kh) * 16 + ml] = acc[v];
  }
  __syncthreads();
  for (int o = tid; o < 256; o += 128) {
    const float s = pt[o] + pt[256 + o] + pt[512 + o] + pt[768 + o];
    const int ky = o >> 4, n = o & 15;
    t2[(size_t)blockIdx.x * 256 + ky * 16 + (n & 7) * 2 + (n >> 3)] = s;
  }
}

// ===========================================================================
// K2: per (b,ci, ky-half): x-DFT onto 16 corner kx modes, keep REAL part.
//   xd[bci, kx(16), ky(16), kz(8)] = sum_X t2r*cos + t2i*sin
// ===========================================================================
__global__ __launch_bounds__(128) void k2_fwd_x(const float* __restrict__ t2,
                                                float* __restrict__ xd) {
  __shared__ __attribute__((aligned(16))) float cs[64], sn[64];
  __shared__ __attribute__((aligned(16))) float ax[2048];  // A' 16x128 rowmaj
  __shared__ __attribute__((aligned(16))) float bt[8192];  // B' 128x64 intlv
  const int tid = threadIdx.x;
  const int lane = tid & 31, wave = tid >> 5;
  const int ml = lane & 15, kh = lane >> 4;
  init_tw(cs, sn, tid);
  const int bci = blockIdx.x >> 1;
  const int h = blockIdx.x & 1;  // ky half: 0 -> ky 0..7, 1 -> ky 8..15
  const float* tp = t2 + (size_t)bci * 16384 + h * 128;
  // Coalesced global read of the (X, half) slab, scatter into interleaved B'.
  // global element (X, r): r = kyl*16 + kz*2 + comp  ->  B'(k=X+64*comp,
  //   n=kyl*8+kz) at bt[(((X>>1)+32*comp)*64 + n)*2 + (X&1)]
  for (int i = 0; i < 64; ++i) {
    int g = tid + i * 128;  // 8192
    int X = g >> 7, r = g & 127;
    int n = (r >> 4) * 8 + ((r >> 1) & 7);
    int comp = r & 1;
    bt[(((X >> 1) + 32 * comp) * 64 + n) * 2 + (X & 1)] = tp[X * 256 + r];
  }
  __syncthreads();
  // A': ax[kx, k'] ; k'<64 -> cos(kx*X), else sin(kx*X)
  for (int i = 0; i < 16; ++i) {
    int e = tid + i * 128;  // 2048
    int m = e >> 7, k = e & 127;
    int kxact = (m < 8) ? m : (m + 48);
    ax[e] = (k < 64) ? cs[(kxact * k) & 63] : sn[(kxact * (k - 64)) & 63];
  }
  __syncthreads();

  const int mgl = wave * 16 + ml;  // local mode (B col): ky_l*8 + kz
  v8f acc = {0.f, 0.f, 0.f, 0.f, 0.f, 0.f, 0.f, 0.f};
#pragma unroll
  for (int k0 = 0; k0 < 128; k0 += 4) {
    const int kk = k0 + 2 * kh;
    v2f a = ld2(&ax[ml * 128 + kk]);
    v2f b = ld2(&bt[((kk >> 1) * 64 + mgl) * 2]);
    acc = wmma4(a, b, acc);
  }
#pragma unroll
  for (int v = 0; v < 8; ++v) {
    const int kx = v + 8 * kh;
    xd[(size_t)bci * 2048 + kx * 128 + h * 64 + mgl] = acc[v];
  }
}

// ===========================================================================
// K3: channel mix (weights vary per mode -> plain VALU).
//   om[b,co,mode,{r,i}] = (1/64^3) * sum_ci xd[b,ci,mode] * w{r,i}[ci,co,mode]
// ===========================================================================
__global__ __launch_bounds__(256) void k3_mix(
    const float* __restrict__ xd, const float* __restrict__ w1r,
    const float* __restrict__ w1i, const float* __restrict__ w2r,
    const float* __restrict__ w2i, const float* __restrict__ w3r,
    const float* __restrict__ w3i, const float* __restrict__ w4r,
    const float* __restrict__ w4i, float* __restrict__ om) {
  const int idx = blockIdx.x * 256 + threadIdx.x;  // [0, 131072)
  const int b = idx >> 16;
  const int rem = idx & 65535;
  const int co = rem >> 11;
  const int mode = rem & 2047;
  const int kx = mode >> 7, ky = (mode >> 3) & 15, kz = mode & 7;
  const float *wr, *wi;
  if (kx < 8) {
    if (ky < 8) { wr = w1r; wi = w1i; } else { wr = w3r; wi = w3i; }
  } else {
    if (ky < 8) { wr = w2r; wi = w2i; } else { wr = w4r; wi = w4i; }
  }
  const int wbase = ((co * 8 + (kx & 7)) * 8 + (ky & 7)) * 8 + kz;
  const float* xp = xd + (size_t)b * 65536 + mode;
  float ar = 0.f, ai = 0.f;
#pragma unroll 4
  for (int ci = 0; ci < 32; ++ci) {
    const float v = xp[ci * 2048];
    ar += v * wr[ci * 16384 + wbase];
    ai += v * wi[ci * 16384 + wbase];
  }
  const float sc = 1.0f / 262144.0f;  // 1/(64^3)
  om[(size_t)idx * 2 + 0] = ar * sc;
  om[(size_t)idx * 2 + 1] = ai * sc;
}

// ===========================================================================
// K4: per (b,co): inverse kx -> X expansion (complex, e^{+i}).
//   u[b,co, X(64), mode(128), {r,i}] ; ur = sum or*c - oi*s, ui = or*s + oi*c
// ===========================================================================
__global__ __launch_bounds__(128) void k4_inv_x(const float* __restrict__ om,
                                                float* __restrict__ u) {
  __shared__ __attribute__((aligned(16))) float cs[64], sn[64];
  __shared__ __attribute__((aligned(16))) float lm[4096];  // om[b,co] slab
  __shared__ __attribute__((aligned(16))) float ax[2048];  // A' 64x32 rowmaj
  __shared__ __attribute__((aligned(16))) float bt[8192];  // B' 32x256 intlv
  const int tid = threadIdx.x;
  const int lane = tid & 31, wave = tid >> 5;
  const int ml = lane & 15, kh = lane >> 4;
  init_tw(cs, sn, tid);
  const float* op = om + (size_t)blockIdx.x * 4096;
  for (int i = 0; i < 32; ++i) lm[tid + i * 128] = op[tid + i * 128];
  __syncthreads();
  // A': ax[X, k'] ; k'<16 -> cos(X*kx), else sin(X*kx)
  for (int i = 0; i < 16; ++i) {
    int e = tid + i * 128;  // 2048
    int X = e >> 5, k = e & 31;
    int kq = (k < 16) ? k : (k - 16);
    int kxact = (kq < 8) ? kq : (kq + 48);
    ax[e] = (k < 16) ? cs[(X * kxact) & 63] : sn[(X * kxact) & 63];
  }
  // B': col c = ng*16+n -> (mode = (c>>4)*8+(c&7), comp = (c>>3)&1)
  //   k<16: [or|oi] ; k>=16: comp ? +or : -oi
  for (int i = 0; i < 64; ++i) {
    int e = tid + i * 128;  // 8192
    int pair = e >> 9, rest = e & 511;
    int c = rest >> 1, k = pair * 2 + (rest & 1);
    int mode = (c >> 4) * 8 + (c & 7), comp = (c >> 3) & 1;
    float v;
    if (k < 16) v = lm[(k * 128 + mode) * 2 + comp];
    else if (comp) v = lm[((k - 16) * 128 + mode) * 2 + 0];
    else v = -lm[((k - 16) * 128 + mode) * 2 + 1];
    bt[e] = v;
  }
  __syncthreads();

  float* ub = u + (size_t)blockIdx.x * 16384;
  for (int t = wave; t < 64; t += 4) {  // 4 X-tiles x 16 col-groups
    const int tX = t >> 4, ng = t & 15;
    const int Xa = tX * 16 + ml;
    const int cB = ng * 16 + ml;
    v8f acc = {0.f, 0.f, 0.f, 0.f, 0.f, 0.f, 0.f, 0.f};
#pragma unroll
    for (int k0 = 0; k0 < 32; k0 += 4) {
      const int kk = k0 + 2 * kh;
      v2f a = ld2(&ax[Xa * 32 + kk]);
      v2f b = ld2(&bt[((kk >> 1) * 256 + cB) * 2]);
      acc = wmma4(a, b, acc);
    }
    const int mgB = ng * 8 + (ml & 7);
    const int compB = ml >> 3;
#pragma unroll
    for (int v = 0; v < 8; ++v) {
      const int Xr = tX * 16 + v + 8 * kh;
      ub[Xr * 256 + mgB * 2 + compB] = acc[v];
    }
  }
}

// ===========================================================================
// K5: per (b,co,X): inverse ky -> Y, then kz -> Z with Re extraction.
//   y[Y,Z] = sum_kz vr[Y,kz]*cos(kz*Z) - vi[Y,kz]*sin(kz*Z)
// ===========================================================================
__global__ __launch_bounds__(128) void k5_inv_yz(const float* __restrict__ u,
                                                 float* __restrict__ out) {
  __shared__ __attribute__((aligned(16))) float cs[64], sn[64];
  __shared__ __attribute__((aligned(16))) float lu[256];   // [ky][kz][2]
  __shared__ __attribute__((aligned(16))) float ay[2048];  // A' 64x32 rowmaj
  __shared__ __attribute__((aligned(16))) float ba[512];   // B' 32x16 intlv
  __shared__ __attribute__((aligned(16))) float vt[64 * 16];  // [Y][vr|vi]
  __shared__ __attribute__((aligned(16))) float bz[1024];  // Bz 16x64 intlv
  const int tid = threadIdx.x;
  const int lane = tid & 31, wave = tid >> 5;
  const int ml = lane & 15, kh = lane >> 4;
  init_tw(cs, sn, tid);
  const float* up = u + (size_t)blockIdx.x * 256;
  lu[tid] = up[tid];
  lu[tid + 128] = up[tid + 128];
  __syncthreads();
  // A' (stage A): ay[Y, k'] ; k'<16 -> cos(Y*ky), else sin(Y*ky)
  for (int i = 0; i < 16; ++i) {
    int e = tid + i * 128;  // 2048
    int Y = e >> 5, k = e & 31;
    int kq = (k < 16) ? k : (k - 16);
    int kyact = (kq < 8) ? kq : (kq + 48);
    ay[e] = (k < 16) ? cs[(Y * kyact) & 63] : sn[(Y * kyact) & 63];
  }
  // B' (stage A): col n -> (kz=n&7, comp=n>>3); k<16: [ur|ui];
  //   k>=16: comp ? +ur : -ui
  for (int i = 0; i < 4; ++i) {
    int e = tid + i * 128;  // 512
    int pair = e >> 5, rest = e & 31;
    int n = rest >> 1, k = pair * 2 + (rest & 1);
    int kz = n & 7, comp = n >> 3;
    float v;
    if (k < 16) v = lu[(k * 8 + kz) * 2 + comp];
    else if (comp) v = lu[((k - 16) * 8 + kz) * 2 + 0];
    else v = -lu[((k - 16) * 8 + kz) * 2 + 1];
    ba[e] = v;
  }
  // Bz (stage B): bz[k, Z] ; k<8 -> cos(k*Z), else -sin((k-8)*Z)
  for (int i = 0; i < 8; ++i) {
    int e = tid + i * 128;  // 1024
    int pair = e >> 7, rest = e & 127;
    int Z = rest >> 1, k = pair * 2 + (rest & 1);
    bz[e] = (k < 8) ? cs[(k * Z) & 63] : -sn[((k - 8) * Z) & 63];
  }
  __syncthreads();

  // ---- stage A: v[Y,n], wave w -> Y tile w
  {
    const int y0 = wave * 16;
    v8f acc = {0.f, 0.f, 0.f, 0.f, 0.f, 0.f, 0.f, 0.f};
#pragma unroll
    for (int k0 = 0; k0 < 32; k0 += 4) {
      const int kk = k0 + 2 * kh;
      v2f a = ld2(&ay[(y0 + ml) * 32 + kk]);
      v2f b = ld2(&ba[((kk >> 1) * 16 + ml) * 2]);
      acc = wmma4(a, b, acc);
    }
#pragma unroll
    for (int v = 0; v < 8; ++v) vt[(y0 + v + 8 * kh) * 16 + ml] = acc[v];
  }
  __syncthreads();

  // ---- stage B: y[Y,Z] = vt(A, k-contiguous) x bz
  {
    const int y0 = wave * 16;
    float* ob = out + (size_t)blockIdx.x * 4096;
#pragma unroll
    for (int tZ = 0; tZ < 4; ++tZ) {
      const int Z = tZ * 16 + ml;
      v8f acc = {0.f, 0.f, 0.f, 0.f, 0.f, 0.f, 0.f, 0.f};
#pragma unroll
      for (int k0 = 0; k0 < 16; k0 += 4) {
        const int kk = k0 + 2 * kh;
        v2f a = ld2(&vt[(y0 + ml) * 16 + kk]);
        v2f b = ld2(&bz[((kk >> 1) * 64 + Z) * 2]);
        acc = wmma4(a, b, acc);
      }
#pragma unroll
      for (int v = 0; v < 8; ++v) ob[(y0 + v + 8 * kh) * 64 + Z] = acc[v];
    }
  }
}

// ===========================================================================
extern "C" void kernel_launch(void* const* d_in, const int* in_sizes, int n_in,
                              void* d_out, int out_size, void* d_ws,
                              size_t ws_size, hipStream_t stream) {
  (void)in_sizes; (void)n_in; (void)out_size; (void)ws_size;
  const float* x = (const float*)d_in[0];
  const float* w1r = (const float*)d_in[1];
  const float* w1i = (const float*)d_in[2];
  const float* w2r = (const float*)d_in[3];
  const float* w2i = (const float*)d_in[4];
  const float* w3r = (const float*)d_in[5];
  const float* w3i = (const float*)d_in[6];
  const float* w4r = (const float*)d_in[7];
  const float* w4i = (const float*)d_in[8];
  float* out = (float*)d_out;
  float* ws = (float*)d_ws;

  float* t2 = ws;                    // 2,097,152 floats (8.4 MB)
  float* xd = t2 + 2097152;          //   131,072 floats
  float* om = xd + 131072;           //   262,144 floats
  float* u  = om + 262144;           // 1,048,576 floats  (total ~14.2 MB)

  k1_fwd_yz<<<4096, 128, 0, stream>>>(x, t2);
  k2_fwd_x<<<128, 128, 0, stream>>>(t2, xd);
  k3_mix<<<512, 256, 0, stream>>>(xd, w1r, w1i, w2r, w2i, w3r, w3i, w4r, w4i,
                                  om);
  k4_inv_x<<<64, 128, 0, stream>>>(om, u);
  k5_inv_yz<<<4096, 128, 0, stream>>>(u, out);
}